// Aggregation_53429393162616
// MI455X (gfx1250) — compile-verified
//
#include <hip/hip_runtime.h>
#include <hip/hip_bf16.h>

#define N_NODES 10000
#define N_EDGES 320000
#define FEAT    256

typedef __attribute__((ext_vector_type(2))) float v2f;
typedef __attribute__((ext_vector_type(4))) float v4f;
typedef __attribute__((ext_vector_type(8))) float v8f;

// ---------------------------------------------------------------------------
// Phase 1: scatter-add edge features into seg_sum[N,F] and counts[N] (fp32
// atomics resolved in L2; seg_sum is 10.24 MB and L2-resident on MI455X).
// 64 threads per edge, one float4 (NT load) + 4 global_atomic_add_f32 each.
// A wave covers 512 contiguous bytes of one destination row -> coalesced
// atomic bursts.
// ---------------------------------------------------------------------------
__global__ __launch_bounds__(256) void scatter_mean_accum_kernel(
    const float* __restrict__ src,            // [E, F]
    const long long* __restrict__ edge_index, // [2, E] int64
    float* __restrict__ seg_sum,              // [N, F]
    float* __restrict__ counts)               // [N]
{
    long long gid = (long long)blockIdx.x * blockDim.x + threadIdx.x;
    if (gid >= (long long)N_EDGES * 64) return;
    const int e     = (int)(gid >> 6);
    const int chunk = (int)(gid & 63);        // which float4 of the 256-wide row

    const int dst = (int)edge_index[(long long)N_EDGES + e]; // edge_index[1][e]

    const v4f* srow = (const v4f*)(src + (size_t)e * FEAT);
    const v4f v = __builtin_nontemporal_load(&srow[chunk]);  // stream-once data

    float* base = seg_sum + (size_t)dst * FEAT + chunk * 4;
    atomicAdd(base + 0, v.x);
    atomicAdd(base + 1, v.y);
    atomicAdd(base + 2, v.z);
    atomicAdd(base + 3, v.w);

    if (chunk == 0) atomicAdd(counts + dst, 1.0f);
}

// ---------------------------------------------------------------------------
// Phase 2: out[n,j] = (seg_sum[n,:]/max(count,1), 0 if empty) . W[j,:] + b[j]
// Full-precision fp32 WMMA: V_WMMA_F32_16X16X4_F32, D = A(16x4) x B(4x16) + C.
// One wave owns a 16-row node band x 4 column tiles (64 output columns),
// reusing the A fragment across the 4 WMMAs of each k-step.
//
// A fragment (16x4 f32): lane L -> M = L&15, K = 2*(L>>4)+{0,1}  (2 VGPRs)
// B fragment (4x16 f32): lane L -> N = L&15, K = 2*(L>>4)+{0,1}; B[k][j]=W[j][k]
//   so a row-major float2 load from W row (ct*16+N) at offset k gives it.
// C/D (16x16 f32, 8 VGPRs): lane L -> N = L&15, VGPR v -> M = v + 8*(L>>4)
// ---------------------------------------------------------------------------
__global__ __launch_bounds__(256) void mean_gemm_wmma_kernel(
    const float* __restrict__ seg_sum,   // [N, F]
    const float* __restrict__ counts,    // [N]
    const float* __restrict__ W,         // [F, F] row-major
    const float* __restrict__ bias,      // [F]
    float* __restrict__ out)             // [N, F]
{
    const int lane = threadIdx.x & 31;
    const int wave = threadIdx.x >> 5;
    const int task = blockIdx.x * 8 + wave;    // 8 waves / 256-thread block
    const int nt   = task >> 2;                // node tile: 0..624 (10000/16)
    const int cg   = task & 3;                 // column group: 4 tiles of 16
    if (nt >= N_NODES / 16) return;

    const int m_or_n = lane & 15;              // M for A, N for B/C/D
    const int kh     = (lane >> 4) * 2;        // K sub-offset: 0 or 2

    const int row = nt * 16 + m_or_n;
    const float cnt   = counts[row];
    const float scale = cnt > 0.0f ? 1.0f / cnt : 0.0f; // mean + empty-segment mask

    const float* arow = seg_sum + (size_t)row * FEAT;

    v8f acc[4] = {v8f{}, v8f{}, v8f{}, v8f{}};

    for (int k0 = 0; k0 < FEAT; k0 += 4) {
        // A fragment: two consecutive fp32 of this lane's node row, pre-scaled
        v2f a = *(const v2f*)(arow + k0 + kh);
        a.x *= scale;
        a.y *= scale;
#pragma unroll
        for (int t = 0; t < 4; ++t) {
            const int ct = cg * 4 + t;
            const float* wrow = W + (size_t)(ct * 16 + m_or_n) * FEAT;
            const v2f bf = *(const v2f*)(wrow + k0 + kh);
            // (neg_a, A, neg_b, B, c_mod, C, reuse_a, reuse_b)
            acc[t] = __builtin_amdgcn_wmma_f32_16x16x4_f32(
                false, a, false, bf, (short)0, acc[t], false, false);
        }
    }

#pragma unroll
    for (int t = 0; t < 4; ++t) {
        const int col = (cg * 4 + t) * 16 + m_or_n;  // N for this lane
        const float bv = bias[col];
#pragma unroll
        for (int v = 0; v < 8; ++v) {
            const int m = v + 8 * (lane >> 4);       // C/D VGPR->row mapping
            out[(size_t)(nt * 16 + m) * FEAT + col] = acc[t][v] + bv;
        }
    }
}

// ---------------------------------------------------------------------------
extern "C" void kernel_launch(void* const* d_in, const int* in_sizes, int n_in,
                              void* d_out, int out_size, void* d_ws, size_t ws_size,
                              hipStream_t stream) {
    const float*     src  = (const float*)d_in[0];     // [E, F] fp32
    const long long* eidx = (const long long*)d_in[1]; // [2, E] int64
    // d_in[2] = feature_dim scalar (unused; FEAT is compile-time)
    const float*     W    = (const float*)d_in[3];     // [F, F] fp32
    const float*     bias = (const float*)d_in[4];     // [F]  fp32
    float*           out  = (float*)d_out;             // [N, F] fp32

    float* seg = (float*)d_ws;                         // [N*F]
    float* cnt = seg + (size_t)N_NODES * FEAT;         // [N]

    // Zero accumulators (graph-capture-safe async memset)
    (void)hipMemsetAsync(d_ws, 0,
                         ((size_t)N_NODES * FEAT + N_NODES) * sizeof(float),
                         stream);

    // Phase 1: scatter-add.  E*64 work items, 256 threads/block (8 wave32)
    const long long work = (long long)N_EDGES * 64;
    const int blocks1 = (int)((work + 255) / 256);
    scatter_mean_accum_kernel<<<blocks1, 256, 0, stream>>>(src, eidx, seg, cnt);

    // Phase 2: mean + fp32 WMMA GEMM.  625 node tiles x 4 col groups = 2500
    // wave-tasks, 8 waves per block.
    const int tasks = (N_NODES / 16) * 4;
    const int blocks2 = (tasks + 7) / 8;
    mean_gemm_wmma_kernel<<<blocks2, 256, 0, stream>>>(seg, cnt, W, bias, out);
}